// SparseConvBlock_56392920596826
// MI455X (gfx1250) — compile-verified
//
#include <hip/hip_runtime.h>
#include <hip/hip_bf16.h>

#define NVOX 200000
#define KOFF 27
#define PP   100000
#define CIN  64
#define COUT 64
#define BN_EPS 1e-5f

typedef __attribute__((ext_vector_type(2))) float v2f;
typedef __attribute__((ext_vector_type(8))) float v8f;

// ---------------------------------------------------------------------------
// Phase 1: sparse conv  gather -> WMMA f32 16x16x4 GEMM -> atomic scatter-add
// One wave handles 16 rulebook pairs producing a 16x64 partial tile.
//
// A-matrix (16x4 f32) VGPR layout (ISA 7.12.2): lanes 0-15 hold K={0,1},
// lanes 16-31 hold K={2,3} for row M = lane&15.  B mirrors it: lanes 0-15
// supply rows {0,1}, lanes 16-31 rows {2,3}, column N = lane&15.
//
// LDS weight layout is K-pair interleaved: Wk[(r>>1)*128 + c*2 + (r&1)]
// so a B fragment (two consecutive K rows, one column) is one aligned
// 8-byte ds_load_b64 — no register repacking before v_wmma.
// ---------------------------------------------------------------------------
__global__ __launch_bounds__(256) void sparse_conv_scatter(
    const float* __restrict__ feat,     // [NVOX, CIN]
    const float* __restrict__ weight,   // [KOFF, CIN, COUT]
    const int*   __restrict__ in_idx,   // [KOFF, PP]
    const int*   __restrict__ out_idx,  // [KOFF, PP]
    float*       __restrict__ out)      // [NVOX, COUT] accumulator
{
  __shared__ float Wk[CIN * COUT];      // 16 KB, pair-interleaved layout

  const int k = blockIdx.y;
  const float* wsrc = weight + (size_t)k * CIN * COUT;
  for (int i = threadIdx.x; i < CIN * COUT; i += 256) {
    const int r = i >> 6;               // K row
    const int c = i & 63;               // N column
    Wk[(r >> 1) * 128 + c * 2 + (r & 1)] = wsrc[i];
  }
  __syncthreads();

  const int lane  = threadIdx.x & 31;
  const int wave  = threadIdx.x >> 5;
  const int tile  = blockIdx.x * 8 + wave;   // 16-pair tile
  const int p0    = tile * 16;
  if (p0 >= PP) return;                      // PP % 16 == 0, tile fully valid

  const int m     = lane & 15;               // pair row / B column this lane owns
  const int khalf = lane >> 4;               // 0: K cols {0,1}, 1: K cols {2,3}

  // gather source row for A fragments
  const int arow = in_idx[(size_t)k * PP + p0 + m];
  const float* src = feat + (size_t)arow * CIN + khalf * 2;

  // Preload all 16 A fragments (covers the 16x64 gathered tile exactly once)
  v2f a[16];
#pragma unroll
  for (int kk = 0; kk < 16; ++kk)
    a[kk] = *(const v2f*)(src + kk * 4);

  // Per-lane LDS base: fragment(kk,nb) at bbase + kk*256 + nb*32 floats
  const float* bbase = &Wk[khalf * 128 + m * 2];

  v8f acc[4] = {v8f{0.f}, v8f{0.f}, v8f{0.f}, v8f{0.f}};

#pragma unroll
  for (int kk = 0; kk < 16; ++kk) {
#pragma unroll
    for (int nb = 0; nb < 4; ++nb) {
      v2f b = *(const v2f*)(bbase + kk * 256 + nb * 32);  // single ds_load_b64
      acc[nb] = __builtin_amdgcn_wmma_f32_16x16x4_f32(
          /*neg_a=*/false, a[kk], /*neg_b=*/false, b,
          /*c_mod=*/(short)0, acc[nb], /*reuse_a=*/false, /*reuse_b=*/false);
    }
  }

  // C/D layout: lane n = lane&15; VGPR j -> row M = j + 8*(lane>>4)
  const int n     = lane & 15;
  const int mbase = (lane >> 4) * 8;
#pragma unroll
  for (int j = 0; j < 8; ++j) {
    const int orow = out_idx[(size_t)k * PP + p0 + mbase + j];
    float* dst = out + (size_t)orow * COUT + n;
#pragma unroll
    for (int nb = 0; nb < 4; ++nb)
      atomicAdd(dst + nb * 16, acc[nb][j]);
  }
}

// ---------------------------------------------------------------------------
// Phase 2a: per-channel sum / sumsq of (accum + bias) over all NVOX rows.
// Thread t covers channel t&63, row group t>>6; block covers STAT_ROWS rows.
// ---------------------------------------------------------------------------
#define STAT_ROWS 512

__global__ __launch_bounds__(256) void bn_stats(
    const float* __restrict__ out,      // [NVOX, COUT] conv accumulator
    const float* __restrict__ bias,     // [COUT]
    float*       __restrict__ sums)     // [128]: sum[64], sumsq[64]
{
  const int c  = threadIdx.x & 63;
  const int rg = threadIdx.x >> 6;     // 0..3
  const float b = bias[c];

  const int r_begin = blockIdx.x * STAT_ROWS;
  const int r_end   = (r_begin + STAT_ROWS < NVOX) ? (r_begin + STAT_ROWS) : NVOX;

  float s = 0.f, s2 = 0.f;
  for (int r = r_begin + rg; r < r_end; r += 4) {
    float x = out[(size_t)r * COUT + c] + b;
    s  += x;
    s2 += x * x;
  }

  __shared__ float red[256];
  red[threadIdx.x] = s;
  __syncthreads();
  if (rg == 0) {
    float t = red[c] + red[64 + c] + red[128 + c] + red[192 + c];
    atomicAdd(&sums[c], t);
  }
  __syncthreads();
  red[threadIdx.x] = s2;
  __syncthreads();
  if (rg == 0) {
    float t = red[c] + red[64 + c] + red[128 + c] + red[192 + c];
    atomicAdd(&sums[64 + c], t);
  }
}

// ---------------------------------------------------------------------------
// Phase 2b: in-place bias + batchnorm (batch stats, biased var) + ReLU.
// ---------------------------------------------------------------------------
__global__ __launch_bounds__(256) void bn_finalize(
    float*       __restrict__ out,      // [NVOX, COUT] in-place
    const float* __restrict__ bias,
    const float* __restrict__ gamma,
    const float* __restrict__ beta,
    const float* __restrict__ sums)     // [128]
{
  const size_t total = (size_t)NVOX * COUT;
  size_t idx = (size_t)blockIdx.x * blockDim.x + threadIdx.x;
  if (idx >= total) return;
  const int c = (int)(idx & 63);

  const float inv_n = 1.0f / (float)NVOX;
  const float mean  = sums[c] * inv_n;
  const float var   = sums[64 + c] * inv_n - mean * mean;
  const float scale = gamma[c] * rsqrtf(var + BN_EPS);

  float x = out[idx] + bias[c];
  float y = (x - mean) * scale + beta[c];
  out[idx] = fmaxf(y, 0.0f);
}

// ---------------------------------------------------------------------------
extern "C" void kernel_launch(void* const* d_in, const int* in_sizes, int n_in,
                              void* d_out, int out_size, void* d_ws, size_t ws_size,
                              hipStream_t stream) {
  const float* feat   = (const float*)d_in[0];   // [NVOX, CIN]
  const float* weight = (const float*)d_in[1];   // [KOFF, CIN, COUT]
  const float* bias   = (const float*)d_in[2];   // [COUT]
  const float* gamma  = (const float*)d_in[3];   // [COUT]
  const float* beta   = (const float*)d_in[4];   // [COUT]
  const int*   in_idx = (const int*)d_in[5];     // [KOFF, PP]
  const int*   out_idx= (const int*)d_in[6];     // [KOFF, PP]
  float* out = (float*)d_out;                    // [NVOX, COUT]
  float* sums = (float*)d_ws;                    // 128 floats

  // Zero the accumulator (d_out) and the stats buffer (stream ops: capture-safe)
  hipMemsetAsync(out, 0, (size_t)NVOX * COUT * sizeof(float), stream);
  hipMemsetAsync(sums, 0, 128 * sizeof(float), stream);

  // Phase 1: gather -> WMMA GEMM -> scatter-add
  {
    const int tiles_per_k = PP / 16;                     // 6250
    dim3 grid((tiles_per_k + 7) / 8, KOFF);              // 8 waves per block
    sparse_conv_scatter<<<grid, 256, 0, stream>>>(feat, weight, in_idx, out_idx, out);
  }

  // Phase 2a: channel statistics
  {
    dim3 grid((NVOX + STAT_ROWS - 1) / STAT_ROWS);
    bn_stats<<<grid, 256, 0, stream>>>(out, bias, sums);
  }

  // Phase 2b: bias + BN + ReLU in place
  {
    const size_t total = (size_t)NVOX * COUT;
    dim3 grid((unsigned)((total + 255) / 256));
    bn_finalize<<<grid, 256, 0, stream>>>(out, bias, gamma, beta, sums);
  }
}